// GNNEncoder_2018634629676
// MI455X (gfx1250) — compile-verified
//
#include <hip/hip_runtime.h>
#include <hip/hip_bf16.h>

// GraphSAGE 2-layer forward for MI455X (gfx1250, wave32).
// GEMMs use V_WMMA_F32_16X16X4_F32 (native f32 matrix pipe, no precision loss).

typedef __attribute__((ext_vector_type(2))) float v2f;
typedef __attribute__((ext_vector_type(8))) float v8f;

#define D_IN  128
#define D_HID 256

// ---------------------------------------------------------------- zero fill
__global__ void k_zero_f4(float4* __restrict__ p, long long n4) {
    long long i = (long long)blockIdx.x * blockDim.x + threadIdx.x;
    long long stride = (long long)gridDim.x * blockDim.x;
    float4 z = make_float4(0.f, 0.f, 0.f, 0.f);
    for (; i < n4; i += stride) p[i] = z;
}

// ---------------------------------------------------------------- in-degree
__global__ void k_degree(const int* __restrict__ dst, float* __restrict__ deg,
                         int nEdges) {
    int e = blockIdx.x * blockDim.x + threadIdx.x;
    if (e < nEdges) atomicAdd(&deg[dst[e]], 1.0f);
}

__global__ void k_recip(float* __restrict__ deg, int n) {
    int i = blockIdx.x * blockDim.x + threadIdx.x;
    if (i < n) {
        float d = deg[i];
        deg[i] = (d > 0.f) ? (1.0f / d) : 0.0f;   // == where(deg>0, 1/max(deg,1), 0)
    }
}

// --------------------------------------------- edge scatter-add (gather + atomic)
// thread = one (edge, channel); channel index is the fast dimension -> coalesced
// gather of feat[src] rows; f32 atomic adds resolved at L2.
__global__ void k_scatter_add(const float* __restrict__ feat,
                              const int* __restrict__ src,
                              const int* __restrict__ dst,
                              float* __restrict__ agg,
                              int nEdges, int dShift) {
    long long tid   = (long long)blockIdx.x * blockDim.x + threadIdx.x;
    long long total = (long long)nEdges << dShift;
    if (tid >= total) return;
    int e = (int)(tid >> dShift);
    int c = (int)(tid & ((1 << dShift) - 1));
    int s = src[e];
    int d = dst[e];
    float v = feat[((size_t)s << dShift) + c];
    atomicAdd(&agg[((size_t)d << dShift) + c], v);
}

// ---------------------------------------------------------------- mean scale
__global__ void k_scale_rows(float* __restrict__ agg,
                             const float* __restrict__ deg_inv,
                             int nNodes, int dShift) {
    long long tid   = (long long)blockIdx.x * blockDim.x + threadIdx.x;
    long long total = (long long)nNodes << dShift;
    if (tid >= total) return;
    int node = (int)(tid >> dShift);
    agg[tid] *= deg_inv[node];
}

// ------------------------------------------------- fused dual-GEMM + bias + ReLU
// out[r, :] = relu( Amean[r,:] @ Wl + Aself[r,:] @ Wr + bias ),  out is [rows x 256]
// One wave per 16-row M-tile; wave keeps all 16 N-tiles (16x v8f = 128 VGPRs) and
// streams K in steps of 4 through V_WMMA_F32_16X16X4_F32.
//
// A 16x4 f32 layout (ISA 7.12.2): lane = M (mod 16); VGPR0/1 hold K = kb, kb+1
// with kb = 2*(lane/16).  B 4x16 mirrors it with lane = N.  C/D 16x16: VGPR v
// holds M = v + 8*(lane/16), N = lane%16.
__global__ __launch_bounds__(128) void k_sage_gemm_relu(
    const float* __restrict__ Amean,   // [rows][K]  (already mean-scaled)
    const float* __restrict__ Aself,   // [rows][K]
    const float* __restrict__ Wl,      // [K][256]
    const float* __restrict__ Wr,      // [K][256]
    const float* __restrict__ bias,    // [256]
    float* __restrict__ out,           // [rows][256]
    int rows, int K) {
    const int NT = D_HID / 16;                    // 16 N-tiles
    int tid   = threadIdx.x;
    int wave  = tid >> 5;
    int lane  = tid & 31;
    int mtile = blockIdx.x * 4 + wave;
    int row0  = mtile << 4;
    if (row0 >= rows) return;

    int r  = lane & 15;            // M index for A, N index for B/C/D
    int kb = (lane >> 4) << 1;     // 0 or 2: which K pair this half-wave holds
    int mh = (lane >> 4) << 3;     // 0 or 8: M offset for C/D halves

    const float* aMrow = Amean + (size_t)(row0 + r) * K;
    const float* aSrow = Aself + (size_t)(row0 + r) * K;

    v8f acc[NT];
#pragma unroll
    for (int nt = 0; nt < NT; ++nt) {
        float bv = bias[nt * 16 + r];
        v8f t = {bv, bv, bv, bv, bv, bv, bv, bv};
        acc[nt] = t;
    }

    for (int k0 = 0; k0 < K; k0 += 4) {
        v2f aM = *(const v2f*)(aMrow + k0 + kb);   // K = k0+kb, k0+kb+1
        v2f aS = *(const v2f*)(aSrow + k0 + kb);
        const float* wl0 = Wl + (size_t)(k0 + kb) * D_HID + r;
        const float* wr0 = Wr + (size_t)(k0 + kb) * D_HID + r;
#pragma unroll
        for (int nt = 0; nt < NT; ++nt) {
            v2f bL;
            bL.x = wl0[nt * 16];
            bL.y = wl0[D_HID + nt * 16];
            acc[nt] = __builtin_amdgcn_wmma_f32_16x16x4_f32(
                false, aM, false, bL, (short)0, acc[nt], false, false);
            v2f bR;
            bR.x = wr0[nt * 16];
            bR.y = wr0[D_HID + nt * 16];
            acc[nt] = __builtin_amdgcn_wmma_f32_16x16x4_f32(
                false, aS, false, bR, (short)0, acc[nt], false, false);
        }
    }

#pragma unroll
    for (int nt = 0; nt < NT; ++nt) {
#pragma unroll
        for (int v = 0; v < 8; ++v) {
            float val = acc[nt][v];
            val = val > 0.f ? val : 0.f;           // ReLU
            out[(size_t)(row0 + mh + v) * D_HID + nt * 16 + r] = val;
        }
    }
}

// ---------------------------------------------------------------------- launch
extern "C" void kernel_launch(void* const* d_in, const int* in_sizes, int n_in,
                              void* d_out, int out_size, void* d_ws, size_t ws_size,
                              hipStream_t stream) {
    const float* x    = (const float*)d_in[0];          // [N,128]
    const int*   ei   = (const int*)d_in[1];            // [2,E]
    const float* W1l  = (const float*)d_in[2];          // [128,256]
    const float* W1r  = (const float*)d_in[3];
    const float* b1   = (const float*)d_in[4];          // [256]
    const float* W2l  = (const float*)d_in[5];          // [256,256]
    const float* W2r  = (const float*)d_in[6];
    const float* b2   = (const float*)d_in[7];
    float*       out  = (float*)d_out;                  // [N,256]

    const int N = in_sizes[0] / D_IN;                   // 50000
    const int E = in_sizes[1] / 2;                      // 800000
    const int* src = ei;
    const int* dst = ei + E;

    // workspace layout (floats): [deg_inv N][agg1 N*128][agg2 N*256][h N*256]
    float* deg  = (float*)d_ws;
    float* agg1 = deg  + (size_t)N;
    float* agg2 = agg1 + (size_t)N * D_IN;
    float* h    = agg2 + (size_t)N * D_HID;

    const int BLK = 256;

    // 1) zero deg + agg1 + agg2 (contiguous region)
    {
        long long zf = (long long)N * (1 + D_IN + D_HID);
        long long n4 = zf / 4;                           // N%4==0 -> exact
        int grid = (int)((n4 + BLK - 1) / BLK);
        if (grid > 65535 * 8) grid = 65535 * 8;          // grid-stride covers rest
        k_zero_f4<<<grid, BLK, 0, stream>>>((float4*)deg, n4);
    }

    // 2) degree + reciprocal
    k_degree<<<(E + BLK - 1) / BLK, BLK, 0, stream>>>(dst, deg, E);

    // 3) layer-1 scatter-add of x into agg1 (can overlap deg; recip comes after)
    {
        long long total = (long long)E * D_IN;
        k_scatter_add<<<(int)((total + BLK - 1) / BLK), BLK, 0, stream>>>(
            x, src, dst, agg1, E, 7);
    }

    k_recip<<<(N + BLK - 1) / BLK, BLK, 0, stream>>>(deg, N);

    // 4) mean1 = agg1 * deg_inv
    {
        long long total = (long long)N * D_IN;
        k_scale_rows<<<(int)((total + BLK - 1) / BLK), BLK, 0, stream>>>(
            agg1, deg, N, 7);
    }

    // 5) h = relu(mean1 @ W1_l + x @ W1_r + b1)
    {
        int mtiles = (N + 15) / 16;
        k_sage_gemm_relu<<<(mtiles + 3) / 4, 128, 0, stream>>>(
            agg1, x, W1l, W1r, b1, h, N, D_IN);
    }

    // 6) layer-2 scatter-add of h into agg2
    {
        long long total = (long long)E * D_HID;
        k_scatter_add<<<(int)((total + BLK - 1) / BLK), BLK, 0, stream>>>(
            h, src, dst, agg2, E, 8);
    }

    // 7) mean2 = agg2 * deg_inv
    {
        long long total = (long long)N * D_HID;
        k_scale_rows<<<(int)((total + BLK - 1) / BLK), BLK, 0, stream>>>(
            agg2, deg, N, 8);
    }

    // 8) out = relu(mean2 @ W2_l + h @ W2_r + b2)
    {
        int mtiles = (N + 15) / 16;
        k_sage_gemm_relu<<<(mtiles + 3) / 4, 128, 0, stream>>>(
            agg2, h, W2l, W2r, b2, out, N, D_HID);
    }
}